// KGNN_head_71322226917907
// MI455X (gfx1250) — compile-verified
//
#include <hip/hip_runtime.h>

// KGNN neighbor-aggregation head, fp32, gfx1250 (MI455X).
// One wave32 handles 2 groups: scores via V_WMMA_F32_16X16X4_F32 (fp32-exact),
// weighted entity sum via coalesced float4 VALU FMAs.
//
// Roofline: 576 MB traffic @ 23.3 TB/s ~= 25 us floor; compute (even with the
// 16x WMMA lane waste) needs < 100 TF/s f32 -- far under the matrix ceiling,
// so the design goal is purely clean streaming loads + enough waves in flight.

typedef float v2f __attribute__((ext_vector_type(2)));
typedef float v8f __attribute__((ext_vector_type(8)));

#define NSHAPE 64
#define DIM    64
#define NNB    8

__global__ __launch_bounds__(256) void kgnn_head_kernel(
    const float* __restrict__ drug,   // [B, 64, 64]
    const float* __restrict__ rel,    // [B, 512, 64]
    const float* __restrict__ ent,    // [B, 512, 64]
    float* __restrict__ out,          // [B, 64, 64]
    int npairs)
{
  const int lane = threadIdx.x & 31;
  const int wave = threadIdx.x >> 5;
  const int p    = blockIdx.x * 8 + wave;        // pair index: 2 groups per wave
  if (p >= npairs) return;                       // wave-uniform guard

  const int b  = p >> 5;                         // 32 pairs per batch element
  const int g0 = (p & 31) * 2;

  const float* drug_base = drug + ((size_t)b * NSHAPE + g0) * DIM;             // 2 x 64
  const float* rel_base  = rel  + ((size_t)(b * NSHAPE + g0) * NNB) * DIM;     // 16 x 64
  const float* ent_base  = ent  + ((size_t)(b * NSHAPE + g0) * NNB) * DIM;     // 16 x 64
  float*       out_base  = out  + ((size_t)b * NSHAPE + g0) * DIM;             // 2 x 64

  const int r = lane & 15;   // A row / B column index
  const int h = lane >> 4;   // half-wave: selects K sub-pair {2h, 2h+1}

  // Prefetch the 4KB ent tile (64 cachelines; 2 x 32 lanes x 64B stride).
  __builtin_prefetch(ent_base + lane * 16, 0, 1);
  __builtin_prefetch(ent_base + 512 + lane * 16, 0, 1);

  // --- Phase 1: scores via fp32 WMMA, D = rel_tile(16x64) x drugcols(64x16) ---
  // B columns replicate the two drug vectors: column c -> drug[g0 + (c&1)].
  // Only D[*,0] and D[*,1] are consumed, so no masking of columns 2..15 is
  // needed -- every lane's B load is unconditional (no exec-mask branches).
  const float* arow = rel_base + r * DIM;
  const float* dsel = drug_base + (r & 1) * DIM;

  v8f acc = {};
#pragma unroll
  for (int kk = 0; kk < 16; ++kk) {
    const int k0 = kk * 4 + 2 * h;
    // A (16x4 f32 layout): lane(r,h) holds A[r, 2h], A[r, 2h+1]
    v2f a  = *(const v2f*)(arow + k0);
    // B (4x16 f32 layout): lane(c=r,h) holds B[2h, c], B[2h+1, c]
    v2f bm = *(const v2f*)(dsel + k0);
    acc = __builtin_amdgcn_wmma_f32_16x16x4_f32(false, a, false, bm,
                                                (short)0, acc, false, false);
  }

  // D[m,n] layout: vgpr = m&7, lane = n + 16*(m>>3).
  //   score(g0, j) = D[j,   0] lives in acc[j] @ lane 0
  //   score(g1, j) = D[j+8, 1] lives in acc[j] @ lane 17
  // Lanes 0-15 need g0's scores, lanes 16-31 need g1's: one indexed shuffle.
  const int src = h * 17;
  float s[8];
#pragma unroll
  for (int j = 0; j < 8; ++j) s[j] = __shfl(acc[j], src, 32);

  // --- Phase 2: out[g,:] = sum_n score[g,n] * ent[g,n,:]  (coalesced float4) ---
  const int    gsel   = h;                 // lanes 0-15 -> g0, 16-31 -> g1
  const int    dstart = r * 4;             // each half-wave covers all 64 d's
  const float* erow   = ent_base + gsel * (NNB * DIM) + dstart;

  float4 o = make_float4(0.f, 0.f, 0.f, 0.f);
#pragma unroll
  for (int n = 0; n < 8; ++n) {
    const float4 e = *(const float4*)(erow + n * DIM);
    o.x += s[n] * e.x;
    o.y += s[n] * e.y;
    o.z += s[n] * e.z;
    o.w += s[n] * e.w;
  }
  *(float4*)(out_base + gsel * DIM + dstart) = o;
}

extern "C" void kernel_launch(void* const* d_in, const int* in_sizes, int n_in,
                              void* d_out, int out_size, void* d_ws, size_t ws_size,
                              hipStream_t stream) {
  const float* drug = (const float*)d_in[0];
  const float* rel  = (const float*)d_in[1];
  const float* ent  = (const float*)d_in[2];
  float*       out  = (float*)d_out;

  const int Bsz    = in_sizes[0] / (NSHAPE * DIM);   // 2048
  const int npairs = Bsz * (NSHAPE / 2);             // 65536 waves of work
  const int blocks = (npairs + 7) / 8;               // 8 waves per 256-thread block

  kgnn_head_kernel<<<dim3(blocks), dim3(256), 0, stream>>>(drug, rel, ent, out, npairs);
}